// HyperbolicLMHeadV2_91001767068238
// MI455X (gfx1250) — compile-verified
//
#include <hip/hip_runtime.h>
#include <hip/hip_bf16.h>
#include <math.h>

// Problem dims (fixed by reference): M = B*L = 1024, N = V = 50257, K = 768 (+1 time dim)
#define M_TOTAL 1024
#define N_TOTAL 50257
#define K_TOTAL 768
#define K_IN    769
// GEMM tiling
#define BM 128
#define BN 64
#define KB 32
#define LSTR 36      // LDS row stride (floats): 16B-aligned rows (async b128 dest), conflict-free

typedef float v2f __attribute__((ext_vector_type(2)));
typedef float v8f __attribute__((ext_vector_type(8)));

__device__ __forceinline__ float sani(float x) {
    if (__builtin_isnan(x)) return 0.0f;
    if (x ==  __builtin_inff()) return  1.0e4f;
    if (x == -__builtin_inff()) return -1.0e4f;
    return x;
}

// ---------------- prep kernels (tiny vs. the GEMM) ----------------

// Writes sanitized spatial part xs[m][k] (if xs != nullptr) and x_t[m] = sqrt(1+||sp||^2)
__global__ __launch_bounds__(256)
void prep_x(const float* __restrict__ hidden, float* __restrict__ xs, float* __restrict__ xt) {
    __shared__ float red[256];
    const int m = blockIdx.x;
    const int t = threadIdx.x;
    const float* row = hidden + (size_t)m * K_IN + 1;
    float s = 0.0f;
    for (int k = t; k < K_TOTAL; k += 256) {
        float v = sani(row[k]);
        if (xs) xs[(size_t)m * K_TOTAL + k] = v;   // uniform branch
        s += v * v;
    }
    red[t] = s; __syncthreads();
    for (int off = 128; off > 0; off >>= 1) {
        if (t < off) red[t] += red[t + off];
        __syncthreads();
    }
    if (t == 0) xt[m] = sqrtf(1.0f + red[0]);
}

// per-vocab-row exp-map scalars: sw[n] = sinh(min(r,3))/max(r,eps), wt[n] = cosh(min(r,3))
__global__ __launch_bounds__(256)
void prep_w_scale(const float* __restrict__ w, float* __restrict__ sw, float* __restrict__ wt) {
    __shared__ float red[256];
    const int n = blockIdx.x;
    const int t = threadIdx.x;
    const float* row = w + (size_t)n * K_TOTAL;
    float s = 0.0f;
    for (int k = t; k < K_TOTAL; k += 256) { float v = row[k]; s += v * v; }
    red[t] = s; __syncthreads();
    for (int off = 128; off > 0; off >>= 1) {
        if (t < off) red[t] += red[t + off];
        __syncthreads();
    }
    if (t == 0) {
        float r  = sqrtf(red[0]);
        float rc = fminf(r, 3.0f);
        sw[n] = sinhf(rc) / fmaxf(r, 1e-8f);
        wt[n] = coshf(rc);
    }
}

// ---------------- main GEMM on RAW weights ----------------
// acc[m][n] = x_sp[m] . w_raw[n]   (f32 WMMA, K=768)
// epilogue: out = sani(tau * (sw[n]*acc - x_t[m]*wt[n]))
// ASYNC_A=true : A tile async-copied from pre-sanitized xs (pure DMA loop, no VGPR staging)
// ASYNC_A=false: A tile register-staged from raw hidden (fallback when workspace is small)

template <bool ASYNC_A>
__global__ __launch_bounds__(256)
void hyper_gemm(const float* __restrict__ asrc, const float* __restrict__ w,
                const float* __restrict__ ls, const float* __restrict__ xt,
                const float* __restrict__ sw, const float* __restrict__ wt,
                float* __restrict__ out) {
    __shared__ float As[2][BM * LSTR];   // 36864 B
    __shared__ float Bs[2][BN * LSTR];   // 18432 B

    const int t    = threadIdx.x;
    const int lane = t & 31;
    const int wid  = t >> 5;
    const int wm   = wid >> 1;            // 0..3  : wave M group (32 rows each)
    const int wn   = wid & 1;             // 0..1  : wave N group (32 cols each)
    const int lrow = lane & 15;           // row within 16x16 tile
    const int lkh  = (lane >> 4) << 1;    // K half: 0 or 2 (A layout: V0=K0/K2, V1=K1/K3)

    const int blockN = blockIdx.x * BN;
    const int blockM = blockIdx.y * BM;

    // staging maps
    const int acol  = t & 31;             // (fallback) A: 32 consecutive K per 32 threads
    const int arow0 = t >> 5;             // (fallback) A rows arow0 + 8*j, j<16
    const int bcol  = (t & 7) << 2;       // 4-float (16B) column chunk
    const int brow0 = t >> 3;             // rows brow0 + 32*j

    v8f acc[2][2];
#pragma unroll
    for (int mt = 0; mt < 2; ++mt)
#pragma unroll
        for (int nt = 0; nt < 2; ++nt)
            acc[mt][nt] = (v8f){0.f,0.f,0.f,0.f,0.f,0.f,0.f,0.f};

    float aReg[16];   // only used by fallback path

    auto gloadA = [&](int kt) {
        const int kbase = kt * KB;
#pragma unroll
        for (int j = 0; j < 16; ++j) {
            const int m = blockM + arow0 + 8 * j;          // always < 1024
            aReg[j] = sani(asrc[(size_t)m * K_IN + 1 + kbase + acol]);
        }
    };
    auto lstoreA = [&](int buf) {
#pragma unroll
        for (int j = 0; j < 16; ++j)
            As[buf][(arow0 + 8 * j) * LSTR + acol] = aReg[j];
    };
    // async copy of pre-sanitized A tile straight into LDS (16B per lane per issue)
    auto asyncA = [&](int kt, int buf) {
        const int kbase = kt * KB;
#pragma unroll
        for (int j = 0; j < 4; ++j) {
            const int row = brow0 + 32 * j;                // 0..127
            const int m   = blockM + row;                  // always < 1024
            const unsigned ldsOff = (unsigned)(uintptr_t)&As[buf][row * LSTR + bcol];
            const unsigned gOff   = (unsigned)(((size_t)m * K_TOTAL + kbase + bcol) * sizeof(float));
            asm volatile("global_load_async_to_lds_b128 %0, %1, %2"
                         :: "v"(ldsOff), "v"(gOff), "s"(asrc)
                         : "memory");
        }
    };
    // async copy of raw weight tile straight into LDS
    auto asyncB = [&](int kt, int buf) {
        const int kbase = kt * KB;
#pragma unroll
        for (int j = 0; j < 2; ++j) {
            int n = blockN + brow0 + 32 * j;
            n = (n < N_TOTAL) ? n : (N_TOTAL - 1);   // clamp: keep EXEC uniform, masked at store
            const unsigned ldsOff = (unsigned)(uintptr_t)&Bs[buf][(brow0 + 32 * j) * LSTR + bcol];
            const unsigned gOff   = (unsigned)(((size_t)n * K_TOTAL + kbase + bcol) * sizeof(float));
            asm volatile("global_load_async_to_lds_b128 %0, %1, %2"
                         :: "v"(ldsOff), "v"(gOff), "s"(w)
                         : "memory");
        }
    };

    // prologue: fill buffer 0
    if constexpr (ASYNC_A) {
        asyncA(0, 0);
    } else {
        gloadA(0);
    }
    asyncB(0, 0);
    if constexpr (!ASYNC_A) lstoreA(0);
    asm volatile("s_wait_asynccnt 0x0" ::: "memory");
    __syncthreads();

    const int KT = K_TOTAL / KB;   // 24
    for (int kt = 0; kt < KT; ++kt) {
        const int buf = kt & 1;
        if (kt + 1 < KT) {
            if constexpr (ASYNC_A) {
                asyncA(kt + 1, buf ^ 1);
            } else {
                gloadA(kt + 1);
            }
            asyncB(kt + 1, buf ^ 1);    // DMA overlaps the WMMA loop below
        }

        const float* Abase = &As[buf][(wm * 32 + lrow) * LSTR + lkh];
        const float* Bbase = &Bs[buf][(wn * 32 + lrow) * LSTR + lkh];
#pragma unroll
        for (int kk = 0; kk < KB; kk += 4) {
            v2f a[2], b[2];
#pragma unroll
            for (int mt = 0; mt < 2; ++mt)
                a[mt] = *(const v2f*)(Abase + mt * 16 * LSTR + kk);
#pragma unroll
            for (int nt = 0; nt < 2; ++nt)
                b[nt] = *(const v2f*)(Bbase + nt * 16 * LSTR + kk);
#pragma unroll
            for (int mt = 0; mt < 2; ++mt)
#pragma unroll
                for (int nt = 0; nt < 2; ++nt)
                    acc[mt][nt] = __builtin_amdgcn_wmma_f32_16x16x4_f32(
                        false, a[mt], false, b[nt], (short)0, acc[mt][nt], false, false);
        }

        if (kt + 1 < KT) {
            if constexpr (!ASYNC_A) lstoreA(buf ^ 1);           // waits A loadcnt here, post-WMMA
            asm volatile("s_wait_asynccnt 0x0" ::: "memory");   // tiles landed in LDS
        }
        __syncthreads();
    }

    // epilogue: out = sani(tau * (sw[n]*acc - x_t[m]*wt[n]))
    const float tau = fminf(fmaxf(ls[0], 0.01f), 2.5f);
#pragma unroll
    for (int mt = 0; mt < 2; ++mt) {
        const int mBase = blockM + wm * 32 + mt * 16 + ((lane >> 4) << 3);
#pragma unroll
        for (int nt = 0; nt < 2; ++nt) {
            const int n = blockN + wn * 32 + nt * 16 + (lane & 15);
            if (n < N_TOTAL) {
                const float wtn = wt[n];
                const float swn = sw[n];
#pragma unroll
                for (int v = 0; v < 8; ++v) {
                    const int m = mBase + v;
                    const float val = tau * (swn * acc[mt][nt][v] - xt[m] * wtn);
                    out[(size_t)m * N_TOTAL + n] = sani(val);
                }
            }
        }
    }
}

// ---------------- launcher ----------------

extern "C" void kernel_launch(void* const* d_in, const int* in_sizes, int n_in,
                              void* d_out, int out_size, void* d_ws, size_t ws_size,
                              hipStream_t stream) {
    const float* hidden = (const float*)d_in[0];   // (2,512,769) f32
    const float* weight = (const float*)d_in[1];   // (50257,768) f32
    const float* lscale = (const float*)d_in[2];   // scalar f32
    float* out = (float*)d_out;                    // (2,512,50257) f32

    // workspace layout (floats): [xs: 1024*768][xt: 1024][sw: 50257][wt: 50257]
    const size_t XS_ELTS   = (size_t)M_TOTAL * K_TOTAL;
    const size_t FULL_NEED = (XS_ELTS + M_TOTAL + 2 * (size_t)N_TOTAL) * sizeof(float);

    float* ws = (float*)d_ws;
    const bool useXs = (ws_size >= FULL_NEED);     // deterministic per run
    float* xs = useXs ? ws : nullptr;
    float* xt = useXs ? (ws + XS_ELTS) : ws;
    float* sw = xt + M_TOTAL;
    float* wt = sw + N_TOTAL;

    prep_x<<<M_TOTAL, 256, 0, stream>>>(hidden, xs, xt);
    prep_w_scale<<<N_TOTAL, 256, 0, stream>>>(weight, sw, wt);

    dim3 grid((N_TOTAL + BN - 1) / BN, M_TOTAL / BM);   // (786, 8)
    if (useXs) {
        hyper_gemm<true><<<grid, 256, 0, stream>>>(xs, weight, lscale, xt, sw, wt, out);
    } else {
        hyper_gemm<false><<<grid, 256, 0, stream>>>(hidden, weight, lscale, xt, sw, wt, out);
    }
}